// SymmetryLinear_6588479832308
// MI455X (gfx1250) — compile-verified
//
#include <hip/hip_runtime.h>
#include <hip/hip_bf16.h>

// -------- CDNA5 (gfx1250) SymmetryLinear --------
// out[b,k,:] = [ind_k, x[b,k,:]] @ W[part(k)] + bias[part(k)]
// Folded:  out[b,k,:] = x[b,k,:] @ W[part(k)][1:33,:]  +  (bias + ind_k*W[p][0,:])
// Per joint: [B x 32] @ [32 x 32] f32 GEMM -> v_wmma_f32_16x16x4_f32, K in 8 chunks of 4.

typedef __attribute__((ext_vector_type(2))) float v2f;
typedef __attribute__((ext_vector_type(8))) float v8f;

// Part mapping from the reference (24 joints -> 15 unique parts, +-1 L/R indicator)
__device__ __constant__ int   c_part_idx[24] = {
    0, 1, 1, 2, 3, 3, 4, 5, 5, 6, 7, 7, 8, 9, 9, 10, 11, 11, 12, 12, 13, 13, 14, 14};
__device__ __constant__ float c_part_ind[24] = {
    0.f,  1.f, -1.f, 0.f,  1.f, -1.f, 0.f,  1.f, -1.f, 0.f,  1.f, -1.f,
    0.f,  1.f, -1.f, 0.f,  1.f, -1.f, 1.f, -1.f,  1.f, -1.f,  1.f, -1.f};

#define ROWS_PER_BLOCK 128
#define J_CNT   24
#define XSTRIDE 768   // J*32 floats per batch row
#define PADK    34    // padded K-stride in LDS (even -> 8B aligned b64, conflict-avoiding)

__global__ __launch_bounds__(256, 4)
void symlin_wmma_kernel(const float* __restrict__ x,
                        const float* __restrict__ weight,   // [15][33][32]
                        const float* __restrict__ bias,     // [1][15][32]
                        float* __restrict__ out)            // [B][24][32]
{
    const int k       = blockIdx.y;                 // joint 0..23
    const int rowBase = blockIdx.x * ROWS_PER_BLOCK;
    const int p       = c_part_idx[k];
    const float ind   = c_part_ind[k];

    __shared__ float sWT[32 * PADK];                // W^T: [n][kk], kk = 0..31
    __shared__ float sX [ROWS_PER_BLOCK * PADK];    // x tile: [m][kk]
    __shared__ float sBias[32];                     // bias_eff

    const int tid = threadIdx.x;                    // 0..255 (8 wave32s)
    const unsigned wbase = (unsigned)p * 1056u;     // p*33*32

    // ---- Stage W^T (32x32) into LDS: sWT[n*PADK + kk] = W[p][kk+1][n]
    #pragma unroll
    for (int i = tid; i < 1024; i += 256) {
        const int n  = i >> 5;
        const int kk = i & 31;
        sWT[n * PADK + kk] = weight[wbase + (unsigned)(kk + 1) * 32u + n];
    }
    // ---- Effective bias: bias[p][n] + ind * W[p][0][n]
    if (tid < 32) {
        sBias[tid] = bias[(unsigned)p * 32u + tid] + ind * weight[wbase + tid];
    }
    // ---- Stage x tile: 128 rows x 32 floats, coalesced float4 (b128) loads
    #pragma unroll
    for (int i = tid; i < 1024; i += 256) {         // 1024 float4s, 4 per thread
        const int m  = i >> 3;                      // row in tile
        const int c4 = i & 7;                       // float4 column
        const float4 v = *reinterpret_cast<const float4*>(
            x + (size_t)(rowBase + m) * XSTRIDE + (size_t)k * 32 + c4 * 4);
        float* dst = &sX[m * PADK + c4 * 4];
        dst[0] = v.x; dst[1] = v.y; dst[2] = v.z; dst[3] = v.w;
    }
    __syncthreads();

    // ---- Per-wave 16x32 output tile via v_wmma_f32_16x16x4_f32
    const int wave  = tid >> 5;       // 0..7 -> rows [wave*16, wave*16+16)
    const int lane  = tid & 31;
    const int halfK = lane >> 4;      // ISA A/B layout: K pair = 2*(lane/16)+{0,1}
    const int l16   = lane & 15;
    const int m     = wave * 16 + l16;

    v8f acc0 = {};   // N = 0..15
    v8f acc1 = {};   // N = 16..31

    #pragma unroll
    for (int kc = 0; kc < 8; ++kc) {
        const int kcol = kc * 4 + 2 * halfK;
        // A 16x4 f32 fragment: lane holds (M=l16, K=kcol..kcol+1)
        const v2f a  = *reinterpret_cast<const v2f*>(&sX[m * PADK + kcol]);
        // B 4x16 f32 fragments: lane holds (N=l16, K=kcol..kcol+1)
        const v2f b0 = *reinterpret_cast<const v2f*>(&sWT[l16 * PADK + kcol]);
        const v2f b1 = *reinterpret_cast<const v2f*>(&sWT[(16 + l16) * PADK + kcol]);
        acc0 = __builtin_amdgcn_wmma_f32_16x16x4_f32(
            false, a, false, b0, (short)0, acc0, false, false);
        acc1 = __builtin_amdgcn_wmma_f32_16x16x4_f32(
            false, a, false, b1, (short)0, acc1, false, false);
    }

    // ---- Store with fused bias. C/D layout: VGPR r -> (M = r + 8*halfK, N = l16)
    const float bv0 = sBias[l16];
    const float bv1 = sBias[16 + l16];
    #pragma unroll
    for (int r = 0; r < 8; ++r) {
        const int row = rowBase + wave * 16 + r + 8 * halfK;
        float* o = out + (size_t)row * XSTRIDE + (size_t)k * 32;
        o[l16]      = acc0[r] + bv0;
        o[16 + l16] = acc1[r] + bv1;
    }
}

extern "C" void kernel_launch(void* const* d_in, const int* in_sizes, int n_in,
                              void* d_out, int out_size, void* d_ws, size_t ws_size,
                              hipStream_t stream) {
    const float* x      = (const float*)d_in[0];   // [131072, 24, 32]
    const float* weight = (const float*)d_in[1];   // [15, 33, 32]
    const float* bias   = (const float*)d_in[2];   // [1, 15, 32]
    float* out          = (float*)d_out;           // [131072, 24, 32]

    const int B = in_sizes[0] / (J_CNT * 32);      // 131072
    dim3 grid(B / ROWS_PER_BLOCK, J_CNT);          // (1024, 24)
    symlin_wmma_kernel<<<grid, 256, 0, stream>>>(x, weight, bias, out);
}